// InformPooling_69200513073193
// MI455X (gfx1250) — compile-verified
//
#include <hip/hip_runtime.h>

// InformPooling for MI455X (gfx1250, wave32).
//
// Time-axis segment mean as WMMA reduction: A = ones(16x4) f32, so
// V_WMMA_F32_16X16X4_F32 computes D[m,n] = sum_k B[k,n] + C[m,n] -> each WMMA
// folds 4 time rows x 16 channels into the accumulator on the XDL pipe.
// All 128 MB of input fits in the 192 MB L2, so segment gathers are
// L2-resident; HBM floor is ~136 MB @ 23.3 TB/s.

typedef __attribute__((ext_vector_type(2))) float v2f;
typedef __attribute__((ext_vector_type(8))) float v8f;

namespace {
constexpr int kB  = 8;
constexpr int kN  = 512;
constexpr int kT0 = 16384, kC0 = 128;
constexpr int kT1 = 8192,  kC1 = 128;
constexpr int kT2 = 4096,  kC2 = 256;
constexpr float kEps = 1e-3f;
}

// Sum rows [s, s+cnt) of one channel column per lane (16 channels per wave,
// lane halves redundant). col points at (row 0, ch_base + lane%16).
__device__ __forceinline__ float seg_sum16(const float* __restrict__ col,
                                           int rowStride, int s, int cnt,
                                           int khalf)
{
  v2f a; a[0] = 1.0f; a[1] = 1.0f;     // all-ones A (16x4)
  v8f acc0 = {};
  v8f acc1 = {};
  const int nquad = cnt > 0 ? (cnt >> 2) : 0;

  // Quad i covers rows s+4i .. s+4i+3; lane-half 0 supplies rows +0,+1 and
  // lane-half 1 supplies rows +2,+3 -> each (VGPR, half) slot holds one full
  // B row, which is all an all-ones-A column sum needs.
  const float* q = col + (size_t)(s + 2 * khalf) * rowStride;
  const size_t quadStep = (size_t)4 * rowStride;

  int i = 0;
  for (; i + 1 < nquad; i += 2) {      // 2 accumulators: break D->C chain
    v2f b0, b1;
    b0[0] = q[0];
    b0[1] = q[rowStride];
    b1[0] = q[quadStep];
    b1[1] = q[quadStep + rowStride];
    acc0 = __builtin_amdgcn_wmma_f32_16x16x4_f32(false, a, false, b0,
                                                 (short)0, acc0, false, false);
    acc1 = __builtin_amdgcn_wmma_f32_16x16x4_f32(false, a, false, b1,
                                                 (short)0, acc1, false, false);
    q += 2 * quadStep;
  }
  if (i < nquad) {
    v2f b0;
    b0[0] = q[0];
    b0[1] = q[rowStride];
    acc0 = __builtin_amdgcn_wmma_f32_16x16x4_f32(false, a, false, b0,
                                                 (short)0, acc0, false, false);
  }

  float total = acc0[0] + acc1[0];     // D[0,n] (or D[8,n]) = column sum
  // <=3 remainder rows, added identically in both lane halves (no divergence
  // that matters: loop bounds are wave-uniform).
  for (int r = s + (nquad << 2); r < s + cnt; ++r)
    total += col[(size_t)r * rowStride];
  return total;
}

__global__ void __launch_bounds__(256)
inform_pooling_kernel(const float* __restrict__ v0,
                      const float* __restrict__ v1,
                      const float* __restrict__ v2,
                      const float* __restrict__ start,
                      const float* __restrict__ duration,
                      float* __restrict__ out)
{
  const int bn = blockIdx.x;           // one block per (b, n)
  const int b  = bn >> 9;
  const int n  = bn & (kN - 1);

  const float st = start[b * kN + n];
  const float en = st + duration[b * kN + n] + kEps;

  const int tid   = threadIdx.x;
  const int wave  = tid >> 5;          // 0..7
  const int lane  = tid & 31;
  const int n16   = lane & 15;
  const int khalf = lane >> 4;

  // Segment bounds (uniform per block -> EXEC all-1s at every WMMA).
  const int s0 = min((int)floorf(st), kT0 - 1);
  const int e0 = min((int)ceilf(en), kT0 - 1);
  const int s1 = min((int)floorf(st * 0.5f), kT1 - 1);
  const int e1 = min((int)ceilf(en * 0.5f), kT1 - 1);
  const int s2 = min((int)floorf(st * 0.25f), kT2 - 1);
  const int e2 = min((int)ceilf(en * 0.25f), kT2 - 1);
  const int c0 = e0 - s0, c1 = e1 - s1, c2 = e2 - s2;

  // Wave w owns output channel groups w*16 (v0), 128+w*16 (v1),
  // 256+w*16 (v2 ch w*16) and 384+w*16 (v2 ch 128+w*16).
  const float* col0  = v0 + (size_t)b * kT0 * kC0 + wave * 16 + n16;
  const float* col1  = v1 + (size_t)b * kT1 * kC1 + wave * 16 + n16;
  const float* col2a = v2 + (size_t)b * kT2 * kC2 + wave * 16 + n16;
  const float* col2b = col2a + 128;

  const float t0  = seg_sum16(col0,  kC0, s0, c0, khalf);
  const float t1  = seg_sum16(col1,  kC1, s1, c1, khalf);
  const float t2a = seg_sum16(col2a, kC2, s2, c2, khalf);
  const float t2b = seg_sum16(col2b, kC2, s2, c2, khalf);

  // Empty range -> NaN in TF -> 0 after nan_to_num; reproduce directly.
  const float m0  = c0 > 0 ? t0  / (float)c0 : 0.0f;
  const float m1  = c1 > 0 ? t1  / (float)c1 : 0.0f;
  const float m2a = c2 > 0 ? t2a / (float)c2 : 0.0f;
  const float m2b = c2 > 0 ? t2b / (float)c2 : 0.0f;

  if (lane < 16) {                     // one lane half carries the 16 channels
    float* o = out + (size_t)bn * 512 + wave * 16 + n16;
    o[0]   = m0;
    o[128] = m1;
    o[256] = m2a;
    o[384] = m2b;
  }
}

extern "C" void kernel_launch(void* const* d_in, const int* in_sizes, int n_in,
                              void* d_out, int out_size, void* d_ws, size_t ws_size,
                              hipStream_t stream) {
  (void)in_sizes; (void)n_in; (void)out_size; (void)d_ws; (void)ws_size;
  const float* v0 = (const float*)d_in[0];
  const float* v1 = (const float*)d_in[1];
  const float* v2 = (const float*)d_in[2];
  const float* st = (const float*)d_in[3];
  const float* du = (const float*)d_in[4];
  float* out = (float*)d_out;

  dim3 grid(kB * kN);   // 4096 blocks, one per (batch, segment)
  dim3 block(256);      // 8 wave32 waves -> all 512 output channels
  hipLaunchKernelGGL(inform_pooling_kernel, grid, block, 0, stream,
                     v0, v1, v2, st, du, out);
}